// PINODSingleGene_45646912421932
// MI455X (gfx1250) — compile-verified
//
#include <hip/hip_runtime.h>
#include <math.h>

// Problem constants from the reference
#define BB 32768   // batch
#define TT 48      // timepoints
#define SS 6       // state dim (2*K)
#define KK 3       // oscillators

typedef __attribute__((ext_vector_type(2))) float v2f;
typedef __attribute__((ext_vector_type(8))) float v8f;

// D = A(16x4) * B(4x16) + C(16x16), fp32 WMMA (CDNA5)
static __device__ __forceinline__ v8f wmma4(v2f a, v2f b, v8f c) {
  return __builtin_amdgcn_wmma_f32_16x16x4_f32(false, a, false, b, (short)0, c,
                                               false, false);
}

// Hardware tanh if the gfx1250 builtin is exposed (V_TANH_F32, TRANS32 op),
// else libm fallback.
static __device__ __forceinline__ float tanh_fast(float x) {
#if __has_builtin(__builtin_amdgcn_tanhf)
  return __builtin_amdgcn_tanhf(x);
#else
  return tanhf(x);
#endif
}

// Exchange value with lane^16 (column-wise half-wave swap).
// Prefer v_permlanex16_b32 with identity selects (pure VALU, no DS pipe);
// fall back to __shfl_xor.
static __device__ __forceinline__ float xhalf(float x) {
#if __has_builtin(__builtin_amdgcn_permlanex16)
  int i = __builtin_bit_cast(int, x);
  i = __builtin_amdgcn_permlanex16(i, i, 0x76543210, 0xfedcba98, false, false);
  return __builtin_bit_cast(float, i);
#else
  return __shfl_xor(x, 16);
#endif
}

// Build a 4x16 B-fragment (rows k0..k0+3 of a 16x16 C-layout tile).
// C-layout: element (m,n) lives at lane (n + 16*(m/8)), vgpr m%8.
// B-layout: lane l needs element (k0 + 2*(l/16) + v, l%16) in b[v].
// => same-lane register if the source half matches our half, else swap halves.
template <int BASE, int SRCHALF>
static __device__ __forceinline__ v2f bfrag(v8f c, int hh) {
  float sx = c[BASE + 2 * SRCHALF + 0];
  float sy = c[BASE + 2 * SRCHALF + 1];
  float ox = xhalf(c[BASE + 2 * (1 - SRCHALF) + 0]);
  float oy = xhalf(c[BASE + 2 * (1 - SRCHALF) + 1]);
  v2f r;
  r.x = (hh == SRCHALF) ? sx : ox;
  r.y = (hh == SRCHALF) ? sy : oy;
  return r;
}

struct OdeK {
  v2f aosc0, aosc1;              // A-frags of Aosc^T (6x6 padded), K=0..7
  v2f aw1_0, aw1_1;              // A-frags of ode_w1^T (16x6 padded to K=8)
  v2f aw2_0, aw2_1, aw2_2, aw2_3;// A-frags of ode_w2^T (6x16), K=0..15
  v8f cb1, cb2;                  // bias broadcast tiles (b1 rows; b2 rows, pad 0)
};

// k = f(z) = Aosc^T*Z + b2 + W2^T * tanh(W1^T*Z + b1)   (all transposed tiles)
static __device__ __forceinline__ v8f feval(v8f z, const OdeK& P, int hh) {
  v2f zb0 = bfrag<0, 0>(z, hh);   // Z rows 0..3
  v2f zb1 = bfrag<4, 0>(z, hh);   // Z rows 4..7 (6,7 are zero)
  v8f h = P.cb1;
  h = wmma4(P.aw1_0, zb0, h);
  h = wmma4(P.aw1_1, zb1, h);
#pragma unroll
  for (int r = 0; r < 8; ++r) h[r] = tanh_fast(h[r]);
  v8f acc = P.cb2;
  acc = wmma4(P.aosc0, zb0, acc);
  acc = wmma4(P.aosc1, zb1, acc);
  v2f t0 = bfrag<0, 0>(h, hh);
  v2f t1 = bfrag<4, 0>(h, hh);
  v2f t2 = bfrag<0, 1>(h, hh);
  v2f t3 = bfrag<4, 1>(h, hh);
  acc = wmma4(P.aw2_0, t0, acc);
  acc = wmma4(P.aw2_1, t1, acc);
  acc = wmma4(P.aw2_2, t2, acc);
  acc = wmma4(P.aw2_3, t3, acc);
  return acc;
}

static __device__ __forceinline__ float gelu_exact(float x) {
  return 0.5f * x * (1.0f + erff(x * 0.7071067811865475f));
}

__global__ __launch_bounds__(128) void pinode_fused_kernel(
    const float* __restrict__ expr, const float* __restrict__ t_eval,
    const float* __restrict__ enc_w1, const float* __restrict__ enc_b1,
    const float* __restrict__ ln_g, const float* __restrict__ ln_b,
    const float* __restrict__ enc_w2, const float* __restrict__ enc_b2,
    const float* __restrict__ enc_w3, const float* __restrict__ enc_b3,
    const float* __restrict__ periods, const float* __restrict__ gammav,
    const float* __restrict__ ode_w1, const float* __restrict__ ode_b1,
    const float* __restrict__ ode_w2, const float* __restrict__ ode_b2,
    const float* __restrict__ dec_w, const float* __restrict__ dec_b,
    const float* __restrict__ baseline, float* __restrict__ out_expr_hat,
    float* __restrict__ out_traj, float* __restrict__ out_amp) {
  const int lane = threadIdx.x & 31;
  const int wv = (blockIdx.x * (blockDim.x >> 5)) + (threadIdx.x >> 5);
  const int b0 = wv << 4;           // first sample of this wave's 16-wide tile
  const int hh = lane >> 4;         // wave half
  const int col = lane & 15;        // sample-within-tile / A-row index

  // ============== Encoder: h = gelu(LN(x@W1+b1)) ; gelu(h@W2+b2) ; @W3+b3 ====
  // Transposed tiles: rows = hidden dims (two 16-row tiles), cols = samples.
  v8f h0, h1;
#pragma unroll
  for (int r = 0; r < 8; ++r) {
    h0[r] = enc_b1[r + 8 * hh];
    h1[r] = enc_b1[16 + r + 8 * hh];
  }
#pragma unroll
  for (int j = 0; j < 12; ++j) {    // K = 48
    const int k = 4 * j + 2 * hh;
    const float2 xv = *(const float2*)(expr + (size_t)(b0 + col) * TT + k);
    v2f bx; bx.x = xv.x; bx.y = xv.y;
    v2f a0 = {enc_w1[k * 32 + col], enc_w1[(k + 1) * 32 + col]};
    v2f a1 = {enc_w1[k * 32 + 16 + col], enc_w1[(k + 1) * 32 + 16 + col]};
    h0 = wmma4(a0, bx, h0);
    h1 = wmma4(a1, bx, h1);
  }
  // LayerNorm over hidden dim (32 rows = 16 vgprs x 2 halves) per sample column
  {
    float part = 0.f;
#pragma unroll
    for (int r = 0; r < 8; ++r) part += h0[r] + h1[r];
    const float mu = (part + xhalf(part)) * (1.0f / 32.0f);
    float p2 = 0.f;
#pragma unroll
    for (int r = 0; r < 8; ++r) {
      float d0 = h0[r] - mu, d1 = h1[r] - mu;
      p2 += d0 * d0 + d1 * d1;
    }
    const float var = (p2 + xhalf(p2)) * (1.0f / 32.0f);
    const float inv = rsqrtf(var + 1e-5f);
#pragma unroll
    for (int r = 0; r < 8; ++r) {
      const int m0 = r + 8 * hh, m1 = 16 + m0;
      h0[r] = gelu_exact((h0[r] - mu) * inv * ln_g[m0] + ln_b[m0]);
      h1[r] = gelu_exact((h1[r] - mu) * inv * ln_g[m1] + ln_b[m1]);
    }
  }
  // Layer 2: 32x32, K = 32 across (h0,h1)
  v8f g0, g1;
#pragma unroll
  for (int r = 0; r < 8; ++r) {
    g0[r] = enc_b2[r + 8 * hh];
    g1[r] = enc_b2[16 + r + 8 * hh];
  }
  {
    auto l2 = [&](int j, v2f bb) {
      const int k = 4 * j + 2 * hh;
      v2f a0 = {enc_w2[k * 32 + col], enc_w2[(k + 1) * 32 + col]};
      v2f a1 = {enc_w2[k * 32 + 16 + col], enc_w2[(k + 1) * 32 + 16 + col]};
      g0 = wmma4(a0, bb, g0);
      g1 = wmma4(a1, bb, g1);
    };
    l2(0, bfrag<0, 0>(h0, hh));
    l2(1, bfrag<4, 0>(h0, hh));
    l2(2, bfrag<0, 1>(h0, hh));
    l2(3, bfrag<4, 1>(h0, hh));
    l2(4, bfrag<0, 0>(h1, hh));
    l2(5, bfrag<4, 0>(h1, hh));
    l2(6, bfrag<0, 1>(h1, hh));
    l2(7, bfrag<4, 1>(h1, hh));
  }
#pragma unroll
  for (int r = 0; r < 8; ++r) {
    g0[r] = gelu_exact(g0[r]);
    g1[r] = gelu_exact(g1[r]);
  }
  // Layer 3: z0^T (6 rows, rest zero), K = 32
  v8f z;
#pragma unroll
  for (int r = 0; r < 8; ++r) {
    const int m = r + 8 * hh;
    z[r] = (m < SS) ? enc_b3[m] : 0.0f;
  }
  {
    auto l3 = [&](int j, v2f bb) {
      const int k = 4 * j + 2 * hh;
      v2f a = {(col < SS) ? enc_w3[k * SS + col] : 0.0f,
               (col < SS) ? enc_w3[(k + 1) * SS + col] : 0.0f};
      z = wmma4(a, bb, z);
    };
    l3(0, bfrag<0, 0>(g0, hh));
    l3(1, bfrag<4, 0>(g0, hh));
    l3(2, bfrag<0, 1>(g0, hh));
    l3(3, bfrag<4, 1>(g0, hh));
    l3(4, bfrag<0, 0>(g1, hh));
    l3(5, bfrag<4, 0>(g1, hh));
    l3(6, bfrag<0, 1>(g1, hh));
    l3(7, bfrag<4, 1>(g1, hh));
  }

  // ============== ODE constants in registers ==============
  float om2[KK], ga[KK];
#pragma unroll
  for (int i = 0; i < KK; ++i) {
    const float om = 6.283185307179586f / periods[i];
    om2[i] = om * om;
    ga[i] = gammav[i];
  }
  // Aosc[p][m]: base = z @ Aosc (p = input state index, m = output index)
  auto oscAt = [&](int p, int m) -> float {
    if (p >= SS || m >= SS) return 0.0f;
    if ((p & 1) && m == p - 1) return 1.0f;
    if (!(p & 1) && m == p + 1) return -om2[p >> 1];
    if ((p & 1) && m == p) return -2.0f * ga[p >> 1];
    return 0.0f;
  };
  OdeK P;
  P.aosc0 = {oscAt(2 * hh, col), oscAt(2 * hh + 1, col)};
  P.aosc1 = {oscAt(4 + 2 * hh, col), oscAt(5 + 2 * hh, col)};
  {
    const int k0 = 2 * hh;
    P.aw1_0 = {ode_w1[k0 * 16 + col], ode_w1[(k0 + 1) * 16 + col]};
    const int k1 = 4 + 2 * hh;
    P.aw1_1 = {(k1 < SS) ? ode_w1[k1 * 16 + col] : 0.0f,
               (k1 + 1 < SS) ? ode_w1[(k1 + 1) * 16 + col] : 0.0f};
  }
  {
    auto w2t = [&](int k) -> float {
      return (col < SS) ? ode_w2[k * SS + col] : 0.0f;
    };
    P.aw2_0 = {w2t(0 + 2 * hh), w2t(1 + 2 * hh)};
    P.aw2_1 = {w2t(4 + 2 * hh), w2t(5 + 2 * hh)};
    P.aw2_2 = {w2t(8 + 2 * hh), w2t(9 + 2 * hh)};
    P.aw2_3 = {w2t(12 + 2 * hh), w2t(13 + 2 * hh)};
  }
#pragma unroll
  for (int r = 0; r < 8; ++r) {
    const int m = r + 8 * hh;
    P.cb1[r] = ode_b1[m];
    P.cb2[r] = (m < SS) ? ode_b2[m] : 0.0f;
  }

  const float dw0 = dec_w[0], dw1 = dec_w[1], dw2 = dec_w[2];
  const float dbias = dec_b[0] + baseline[0];

  // ============== Time integration + streaming outputs ==============
  float a0 = 0.f, a1 = 0.f, a2 = 0.f;  // amplitude accumulators (disp^2 sums)

  auto emit = [&](int t, const v8f& zz) {
    a0 += zz[0] * zz[0];
    a1 += zz[2] * zz[2];
    a2 += zz[4] * zz[4];
    if (lane < 16) {  // all valid state rows live in the lower half-wave
      const int b = b0 + col;
      float* tr = out_traj + ((size_t)t * BB + b) * SS;
      *(float2*)(tr + 0) = make_float2(zz[0], zz[1]);
      *(float2*)(tr + 2) = make_float2(zz[2], zz[3]);
      *(float2*)(tr + 4) = make_float2(zz[4], zz[5]);
      out_expr_hat[(size_t)b * TT + t] =
          zz[0] * dw0 + zz[2] * dw1 + zz[4] * dw2 + dbias;
    }
  };

  emit(0, z);
  for (int i = 0; i < TT - 1; ++i) {
    const float dt = t_eval[i + 1] - t_eval[i];
    const float hs = dt * 0.25f;  // dt / SUBSTEPS
    for (int s = 0; s < 4; ++s) {
      v8f k1 = feval(z, P, hh);
      v8f k2 = feval(z + (0.5f * hs) * k1, P, hh);
      v8f k3 = feval(z + (0.5f * hs) * k2, P, hh);
      v8f k4 = feval(z + hs * k3, P, hh);
      z = z + (hs * (1.0f / 6.0f)) * (k1 + 2.0f * k2 + 2.0f * k3 + k4);
    }
    emit(i + 1, z);
  }

  if (lane < 16) {
    const int b = b0 + col;
    float* ap = out_amp + (size_t)b * KK;
    ap[0] = sqrtf(a0 * (1.0f / (float)TT));
    ap[1] = sqrtf(a1 * (1.0f / (float)TT));
    ap[2] = sqrtf(a2 * (1.0f / (float)TT));
  }
}

extern "C" void kernel_launch(void* const* d_in, const int* in_sizes, int n_in,
                              void* d_out, int out_size, void* d_ws,
                              size_t ws_size, hipStream_t stream) {
  (void)in_sizes; (void)n_in; (void)d_ws; (void)ws_size; (void)out_size;
  const float* expr     = (const float*)d_in[0];
  const float* t_eval   = (const float*)d_in[1];
  const float* enc_w1   = (const float*)d_in[2];
  const float* enc_b1   = (const float*)d_in[3];
  const float* ln_g     = (const float*)d_in[4];
  const float* ln_b     = (const float*)d_in[5];
  const float* enc_w2   = (const float*)d_in[6];
  const float* enc_b2   = (const float*)d_in[7];
  const float* enc_w3   = (const float*)d_in[8];
  const float* enc_b3   = (const float*)d_in[9];
  const float* periods  = (const float*)d_in[10];
  const float* gammav   = (const float*)d_in[11];
  const float* ode_w1   = (const float*)d_in[12];
  const float* ode_b1   = (const float*)d_in[13];
  const float* ode_w2   = (const float*)d_in[14];
  const float* ode_b2   = (const float*)d_in[15];
  const float* dec_w    = (const float*)d_in[16];
  const float* dec_b    = (const float*)d_in[17];
  const float* baseline = (const float*)d_in[18];

  float* out = (float*)d_out;
  float* out_expr_hat = out;                                   // B*T
  float* out_traj     = out + (size_t)BB * TT;                 // T*B*S
  float* out_amp      = out_traj + (size_t)TT * BB * SS;       // B*K

  // 2048 waves of 16 samples each; 4 waves per 128-thread block.
  const int blocks = (BB / 16) / 4;
  pinode_fused_kernel<<<blocks, 128, 0, stream>>>(
      expr, t_eval, enc_w1, enc_b1, ln_g, ln_b, enc_w2, enc_b2, enc_w3, enc_b3,
      periods, gammav, ode_w1, ode_b1, ode_w2, ode_b2, dec_w, dec_b, baseline,
      out_expr_hat, out_traj, out_amp);
}